// CounterfactualODEModel_86655260164528
// MI455X (gfx1250) — compile-verified
//
#include <hip/hip_runtime.h>
#include <math.h>

typedef float v2f __attribute__((ext_vector_type(2)));
typedef float v8f __attribute__((ext_vector_type(8)));
typedef unsigned int u32;
typedef u32 u32x4 __attribute__((ext_vector_type(4)));
typedef int  i32x4 __attribute__((ext_vector_type(4)));
typedef int  i32x8 __attribute__((ext_vector_type(8)));

#define FEAT 32
#define TRT  4
#define HID  64
#define TPTS 100

// ---- LDS layout (one flat block so TDM descriptors get static byte offsets) ----
constexpr int OFF_W1 = 0;                               // (36 x 64)
constexpr int OFF_W2 = OFF_W1 + (FEAT + TRT) * HID;     // (64 x 64)
constexpr int OFF_W3 = OFF_W2 + HID * HID;              // (64 x 32)
constexpr int OFF_B1 = OFF_W3 + HID * FEAT;
constexpr int OFF_B2 = OFF_B1 + HID;
constexpr int OFF_B3 = OFF_B2 + HID;
constexpr int OFF_TR = OFF_B3 + FEAT;                   // 100 x 4 treatments
constexpr int OFF_TS = OFF_TR + TPTS * TRT;             // 100 ts
constexpr int OFF_S  = OFF_TS + TPTS;                   // 40 state vec
constexpr int OFF_H1 = OFF_S + 40;
constexpr int OFF_H2 = OFF_H1 + HID;
constexpr int OFF_F  = OFF_H2 + HID;
constexpr int SMEM_N = OFF_F + FEAT;

#if __has_builtin(__builtin_amdgcn_tensor_load_to_lds) && \
    __has_builtin(__builtin_amdgcn_s_wait_tensorcnt)
#define HAVE_TDM 1
#else
#define HAVE_TDM 0
#endif

#if HAVE_TDM
// 1-D TDM copy: n f32 elements from global -> LDS byte offset.  D# per ISA 8.3/8.4:
// group0: count=1 | lds_addr | global_addr[56:0] | type=2 ("image")
// group1: data_size=4B; tensor_dim0=tile_dim0=n (n<=65535); tensor_dim1=tile_dim1=1
__device__ __forceinline__ void tdm_load_f32(const float* gsrc, u32 lds_byte_off, u32 n)
{
  unsigned long long ga = (unsigned long long)(uintptr_t)gsrc;
  u32x4 g0;
  g0[0] = 1u;                                        // count=1, user descriptor
  g0[1] = lds_byte_off;                              // lds_addr (bytes)
  g0[2] = (u32)(ga & 0xFFFFFFFFull);                 // global_addr[31:0]
  g0[3] = (u32)((ga >> 32) & 0x01FFFFFFull) | (2u << 30);  // addr[56:32] | type=2
  i32x8 g1;
  g1[0] = (int)(2u << 16);                           // wg_mask=0, data_size=2 (4B)
  g1[1] = (int)((n & 0xFFFFu) << 16);                // tensor_dim0[15:0]
  g1[2] = (int)(((n >> 16) & 0xFFFFu) | (1u << 16)); // tensor_dim0[31:16] | tensor_dim1=1
  g1[3] = (int)((n & 0xFFFFu) << 16);                // tile_dim0 = n
  g1[4] = 1;                                         // tile_dim1 = 1
  g1[5] = (int)n;                                    // tensor_dim0_stride = n
  g1[6] = 0;
  g1[7] = 0;
  i32x4 z4 = {0, 0, 0, 0};
#if defined(__clang_major__) && (__clang_major__ >= 23)
  i32x8 z8 = {0, 0, 0, 0, 0, 0, 0, 0};
  __builtin_amdgcn_tensor_load_to_lds(g0, g1, z4, z4, z8, 0);
#else
  __builtin_amdgcn_tensor_load_to_lds(g0, g1, z4, z4, 0);
#endif
}
#endif

// GEMV (1 x K) @ (K x N) on V_WMMA_F32_16X16X4_F32.
// State slice lives in row M=0 of A (lanes 0 / 16 per the 16x4 f32 layout);
// A chunks are hoisted (tile-invariant); K outer / 4 N-tiles inner gives four
// independent accumulate chains between dependent WMMAs.
template<int K, int N, bool TANH>
__device__ __forceinline__ void gemv16(const float* __restrict__ s,
                                       const float* __restrict__ W,
                                       const float* __restrict__ bias,
                                       float* __restrict__ out, int lane)
{
  constexpr int NC = K / 4;     // K chunks of 4
  constexpr int NT = N / 16;    // N tiles of 16
  const int  half = lane >> 4;
  const int  nl   = lane & 15;
  const bool row0 = (nl == 0);

  v2f a[NC];
#pragma unroll
  for (int c = 0; c < NC; ++c) {
    const int ka = 4 * c + 2 * half;
    float sv0 = s[ka];          // unconditional ds_load, then cndmask
    float sv1 = s[ka + 1];
    a[c].x = row0 ? sv0 : 0.0f;
    a[c].y = row0 ? sv1 : 0.0f;
  }

  v8f acc[NT] = {};
#pragma unroll
  for (int c = 0; c < NC; ++c) {
    const int ka = 4 * c + 2 * half;
    v2f b[NT];
#pragma unroll
    for (int t = 0; t < NT; ++t) {
      b[t].x = W[ ka      * N + 16 * t + nl];
      b[t].y = W[(ka + 1) * N + 16 * t + nl];
    }
#pragma unroll
    for (int t = 0; t < NT; ++t)
      acc[t] = __builtin_amdgcn_wmma_f32_16x16x4_f32(
          false, a[c], false, b[t], (short)0, acc[t], false, false);
  }

  if (half == 0) {              // row M=0 of D: acc[t][0], lanes 0-15
#pragma unroll
    for (int t = 0; t < NT; ++t) {
      float r = acc[t][0] + bias[16 * t + nl];
      out[16 * t + nl] = TANH ? tanhf(r) : r;
    }
  }
}

// One evaluation of dx/dt. t is lane-uniform; y is this lane's feature.
__device__ __noinline__ float ode_f(float t, float y, int lane, float* sm)
{
  sm[OFF_S + lane] = y;
  // piecewise-linear treatment interpolation on the uniform ts grid
  float pos = t * (float)(TPTS - 1);
  int idx = (int)floorf(pos) + 1;
  idx = idx < 1 ? 1 : (idx > TPTS - 1 ? TPTS - 1 : idx);
  float t0 = sm[OFF_TS + idx - 1], t1 = sm[OFF_TS + idx];
  float w = (t - t0) / (t1 - t0);
  w = fminf(fmaxf(w, 0.0f), 1.0f);
  if (lane < TRT)
    sm[OFF_S + FEAT + lane] = (1.0f - w) * sm[OFF_TR + (idx - 1) * TRT + lane]
                            +         w  * sm[OFF_TR +  idx      * TRT + lane];
  __syncthreads();
  gemv16<FEAT + TRT, HID, true >(sm + OFF_S,  sm + OFF_W1, sm + OFF_B1, sm + OFF_H1, lane);
  __syncthreads();
  gemv16<HID,        HID, true >(sm + OFF_H1, sm + OFF_W2, sm + OFF_B2, sm + OFF_H2, lane);
  __syncthreads();
  gemv16<HID,       FEAT, false>(sm + OFF_H2, sm + OFF_W3, sm + OFF_B3, sm + OFF_F,  lane);
  __syncthreads();
  float r = sm[OFF_F + lane];
  __syncthreads();
  return r;
}

__global__ void __launch_bounds__(32)
dopri5_neural_ode(const float* __restrict__ x0, const float* __restrict__ treat,
                  const float* __restrict__ ts,
                  const float* __restrict__ W1, const float* __restrict__ b1,
                  const float* __restrict__ W2, const float* __restrict__ b2,
                  const float* __restrict__ W3, const float* __restrict__ b3,
                  float* __restrict__ out)
{
  __shared__ float sm[SMEM_N];
  const int lane = threadIdx.x;

#if HAVE_TDM
  // Big weight matrices via Tensor Data Mover (async), small arrays via VALU
  // loads overlapped with the TDM transfers.
  tdm_load_f32(W1, OFF_W1 * 4, (FEAT + TRT) * HID);
  tdm_load_f32(W2, OFF_W2 * 4, HID * HID);
  tdm_load_f32(W3, OFF_W3 * 4, HID * FEAT);
#else
  for (int i = lane; i < (FEAT + TRT) * HID; i += 32) sm[OFF_W1 + i] = W1[i];
  for (int i = lane; i < HID * HID;          i += 32) sm[OFF_W2 + i] = W2[i];
  for (int i = lane; i < HID * FEAT;         i += 32) sm[OFF_W3 + i] = W3[i];
#endif
  for (int i = lane; i < HID;        i += 32) { sm[OFF_B1 + i] = b1[i]; sm[OFF_B2 + i] = b2[i]; }
  sm[OFF_B3 + lane] = b3[lane];
  for (int i = lane; i < TPTS * TRT; i += 32) sm[OFF_TR + i] = treat[i];
  for (int i = lane; i < TPTS;       i += 32) sm[OFF_TS + i] = ts[i];
#if HAVE_TDM
  __builtin_amdgcn_s_wait_tensorcnt(0);
  asm volatile("" ::: "memory");
#endif
  __syncthreads();

  // Dormand-Prince 5(4) tableau
  const float c2 = 0.2f, c3 = 0.3f, c4 = 0.8f, c5 = 8.0f / 9.0f;
  const float a21 = 0.2f;
  const float a31 = 3.0f / 40.0f, a32 = 9.0f / 40.0f;
  const float a41 = 44.0f / 45.0f, a42 = -56.0f / 15.0f, a43 = 32.0f / 9.0f;
  const float a51 = 19372.0f / 6561.0f, a52 = -25360.0f / 2187.0f,
              a53 = 64448.0f / 6561.0f, a54 = -212.0f / 729.0f;
  const float a61 = 9017.0f / 3168.0f, a62 = -355.0f / 33.0f,
              a63 = 46732.0f / 5247.0f, a64 = 49.0f / 176.0f, a65 = -5103.0f / 18656.0f;
  const float bb1 = 35.0f / 384.0f, bb3 = 500.0f / 1113.0f, bb4 = 125.0f / 192.0f,
              bb5 = -2187.0f / 6784.0f, bb6 = 11.0f / 84.0f;
  const float e1 = 71.0f / 57600.0f, e3 = -71.0f / 16695.0f, e4 = 71.0f / 1920.0f,
              e5 = -17253.0f / 339200.0f, e6 = 22.0f / 525.0f, e7 = -1.0f / 40.0f;
  const float RTOL = 1e-7f, ATOL = 1e-9f;
  const float tend = sm[OFF_TS + TPTS - 1];

  float y = x0[lane];
  out[lane] = y;                        // solution at ts[0]
  int iout = 1;

  float t = sm[OFF_TS + 0];
  float h = 1e-4f;
  float k1 = ode_f(t, y, lane, sm);

  for (int step = 0; step < 20000; ++step) {
    if (!(t < tend - 1e-7f) || iout >= TPTS) break;
    if (t + h >= tend) h = tend - t;

    float ys, k2, k3, k4, k5, k6, k7, ynew;
    ys   = y + h * (a21 * k1);
    k2   = ode_f(t + c2 * h, ys, lane, sm);
    ys   = y + h * (a31 * k1 + a32 * k2);
    k3   = ode_f(t + c3 * h, ys, lane, sm);
    ys   = y + h * (a41 * k1 + a42 * k2 + a43 * k3);
    k4   = ode_f(t + c4 * h, ys, lane, sm);
    ys   = y + h * (a51 * k1 + a52 * k2 + a53 * k3 + a54 * k4);
    k5   = ode_f(t + c5 * h, ys, lane, sm);
    ys   = y + h * (a61 * k1 + a62 * k2 + a63 * k3 + a64 * k4 + a65 * k5);
    k6   = ode_f(t + h,      ys, lane, sm);
    ynew = y + h * (bb1 * k1 + bb3 * k3 + bb4 * k4 + bb5 * k5 + bb6 * k6);
    k7   = ode_f(t + h,    ynew, lane, sm);

    // RMS error norm over 32 features: barrier-free shfl-xor butterfly
    float err = h * (e1 * k1 + e3 * k3 + e4 * k4 + e5 * k5 + e6 * k6 + e7 * k7);
    float sc  = ATOL + RTOL * fmaxf(fabsf(y), fabsf(ynew));
    float r   = err / sc;
    float ss  = r * r;
    ss += __shfl_xor(ss, 16);
    ss += __shfl_xor(ss, 8);
    ss += __shfl_xor(ss, 4);
    ss += __shfl_xor(ss, 2);
    ss += __shfl_xor(ss, 1);
    float en = sqrtf(ss * (1.0f / 32.0f));   // lane-uniform

    const bool nan_step = !(en == en);
    const bool accept   = (!nan_step) && (en <= 1.0f);

    if (accept) {
      float tnew = t + h;
      // dense output: cubic Hermite on [t, tnew] with f0=k1, f1=k7 (FSAL pair)
      while (iout < TPTS && sm[OFF_TS + iout] <= tnew + 1e-7f) {
        float sg = (sm[OFF_TS + iout] - t) / h;
        sg = fminf(fmaxf(sg, 0.0f), 1.0f);
        float s2 = sg * sg, s3 = s2 * sg;
        float h00 =  2.0f * s3 - 3.0f * s2 + 1.0f;
        float h10 =         s3 - 2.0f * s2 + sg;
        float h01 = -2.0f * s3 + 3.0f * s2;
        float h11 =         s3 -        s2;
        out[iout * FEAT + lane] = h00 * y + h10 * h * k1 + h01 * ynew + h11 * h * k7;
        ++iout;
      }
      t = tnew; y = ynew; k1 = k7;   // FSAL
    }

    // step-size controller (safety 0.9, clamp [0.2, 10], NaN -> shrink)
    float fac;
    if (nan_step) {
      fac = 0.2f;
    } else {
      fac = 0.9f * powf(fmaxf(en, 1e-10f), -0.2f);
      fac = fminf(10.0f, fmaxf(0.2f, fac));
    }
    h *= fac;
    h = fmaxf(h, 1e-7f);
  }

  while (iout < TPTS) { out[iout * FEAT + lane] = y; ++iout; }
}

extern "C" void kernel_launch(void* const* d_in, const int* in_sizes, int n_in,
                              void* d_out, int out_size, void* d_ws, size_t ws_size,
                              hipStream_t stream) {
  const float* x0    = (const float*)d_in[0];
  const float* treat = (const float*)d_in[1];
  const float* ts    = (const float*)d_in[2];
  const float* W1    = (const float*)d_in[3];
  const float* b1    = (const float*)d_in[4];
  const float* W2    = (const float*)d_in[5];
  const float* b2    = (const float*)d_in[6];
  const float* W3    = (const float*)d_in[7];
  const float* b3    = (const float*)d_in[8];
  float* out = (float*)d_out;
  dopri5_neural_ode<<<1, 32, 0, stream>>>(x0, treat, ts, W1, b1, W2, b2, W3, b3, out);
}